// GCNConv_torch_sparse_old_58377195487753
// MI455X (gfx1250) — compile-verified
//
#include <hip/hip_runtime.h>
#include <hip/hip_bf16.h>

typedef __attribute__((ext_vector_type(2))) float v2f;
typedef __attribute__((ext_vector_type(8))) float v8f;

#define IN_SIZE 256
#define OUT_SIZE 64

// ---------------- degree / norm kernels ----------------

__global__ __launch_bounds__(256) void k_init_deg(unsigned int* __restrict__ deg, int n) {
  int i = blockIdx.x * 256 + threadIdx.x;
  if (i < n) deg[i] = 0u;
}

__global__ __launch_bounds__(256) void k_count_deg(const int* __restrict__ row,
                                                   const int* __restrict__ col,
                                                   unsigned int* __restrict__ deg, int E) {
  int e = blockIdx.x * 256 + threadIdx.x;
  if (e < E) {
    int r = row[e], c = col[e];
    if (r != c) atomicAdd(&deg[r], 1u);   // native global_atomic_add_u32
  }
}

__global__ __launch_bounds__(256) void k_dinv(const unsigned int* __restrict__ deg,
                                              float* __restrict__ dinv, int n) {
  int i = blockIdx.x * 256 + threadIdx.x;
  if (i < n) dinv[i] = rsqrtf((float)deg[i] + 1.0f);  // deg includes self-loop -> always > 0
}

// ---------------- WMMA f32 GEMM: xw = x @ W.T ----------------
// One wave computes a 16(M) x 64(N) tile. K stepped by 4 (V_WMMA_F32_16X16X4_F32).

__global__ __launch_bounds__(256) void k_gemm(const float* __restrict__ x,
                                              const float* __restrict__ W,
                                              float* __restrict__ xw, int n) {
  const int lane  = threadIdx.x & 31;
  const int wave  = threadIdx.x >> 5;
  const int tile  = blockIdx.x * 8 + wave;          // 16-row M tile
  if (tile * 16 >= n) return;                       // wave-uniform exit: EXEC stays all-1s

  const int ml    = lane & 15;                      // M (for A) / N-in-tile (for B,D)
  const int khalf = (lane >> 4) << 1;               // 0 or 2: upper lanes take K+2,K+3
  int arow = tile * 16 + ml;
  if (arow > n - 1) arow = n - 1;                   // clamp (no EXEC change)
  const float* xrow = x + (size_t)arow * IN_SIZE;

  v8f acc0 = {}, acc1 = {}, acc2 = {}, acc3 = {};
  for (int k = 0; k < IN_SIZE; k += 4) {
    // A: 16x4 f32, lane L holds x[M=L%16][k + 2*(L/16) + {0,1}]
    v2f a = *(const v2f*)(xrow + k + khalf);
    // B: 4x16 f32, B[kk][nn] = W[nn][kk]; lane L holds W[N=L%16 + 16*nt][k + 2*(L/16) + {0,1}]
    const float* wb = W + (size_t)ml * IN_SIZE + k + khalf;
    v2f b0 = *(const v2f*)(wb + (size_t)0 * 16 * IN_SIZE);
    v2f b1 = *(const v2f*)(wb + (size_t)1 * 16 * IN_SIZE);
    v2f b2 = *(const v2f*)(wb + (size_t)2 * 16 * IN_SIZE);
    v2f b3 = *(const v2f*)(wb + (size_t)3 * 16 * IN_SIZE);
    acc0 = __builtin_amdgcn_wmma_f32_16x16x4_f32(false, a, false, b0, (short)0, acc0, false, false);
    acc1 = __builtin_amdgcn_wmma_f32_16x16x4_f32(false, a, false, b1, (short)0, acc1, false, false);
    acc2 = __builtin_amdgcn_wmma_f32_16x16x4_f32(false, a, false, b2, (short)0, acc2, false, false);
    acc3 = __builtin_amdgcn_wmma_f32_16x16x4_f32(false, a, false, b3, (short)0, acc3, false, false);
  }

  // D layout: VGPR v holds D[M = v + 8*(lane/16)][N = lane%16] per 16-col tile
  const int rbase = tile * 16 + ((lane >> 4) << 3);
  float* o = xw + (size_t)rbase * OUT_SIZE + ml;
#pragma unroll
  for (int v = 0; v < 8; ++v) {
    if (rbase + v < n) {
      o[(size_t)v * OUT_SIZE +  0] = acc0[v];
      o[(size_t)v * OUT_SIZE + 16] = acc1[v];
      o[(size_t)v * OUT_SIZE + 32] = acc2[v];
      o[(size_t)v * OUT_SIZE + 48] = acc3[v];
    }
  }
}

// ---------------- out init = self-loop contribution ----------------

__global__ __launch_bounds__(256) void k_self(const float* __restrict__ xw,
                                              const float* __restrict__ dinv,
                                              float* __restrict__ out, int n) {
  size_t idx = (size_t)blockIdx.x * 256 + threadIdx.x;   // over n*64 elements
  if (idx < (size_t)n * OUT_SIZE) {
    int i = (int)(idx >> 6);
    float d = dinv[i];
    out[idx] = d * d * xw[idx];
  }
}

// ---------------- edge scatter: out[r] += dinv[r]*dinv[c]*xw[c] ----------------
// 64 threads per edge (one per output feature); xw is L2-resident (25.6 MB << 192 MB).

__global__ __launch_bounds__(256) void k_scatter(const int* __restrict__ row,
                                                 const int* __restrict__ col,
                                                 const float* __restrict__ dinv,
                                                 const float* __restrict__ xw,
                                                 float* __restrict__ out, int E) {
  int e = blockIdx.x * 4 + (threadIdx.x >> 6);
  int j = threadIdx.x & 63;
  if (e >= E) return;
  int r = row[e];
  int c = col[e];
  if (r == c) return;                               // diagonal dropped by fill_diag
  float norm = dinv[r] * dinv[c];
  unsafeAtomicAdd(out + (size_t)r * OUT_SIZE + j,
                  norm * xw[(size_t)c * OUT_SIZE + j]);  // native global_atomic_add_f32
}

// ---------------- launcher ----------------

extern "C" void kernel_launch(void* const* d_in, const int* in_sizes, int n_in,
                              void* d_out, int out_size, void* d_ws, size_t ws_size,
                              hipStream_t stream) {
  const int*   edge = (const int*)d_in[0];    // (2, E) int32
  const float* x    = (const float*)d_in[1];  // (N, 256) f32
  const float* W    = (const float*)d_in[3];  // (64, 256) f32

  const int E = in_sizes[0] / 2;
  const int n = in_sizes[1] / IN_SIZE;
  const int* row = edge;
  const int* col = edge + E;
  float* out = (float*)d_out;

  // workspace layout: xw [n*64 f32] | deg [n u32] | dinv [n f32]
  float*        xw   = (float*)d_ws;
  unsigned int* deg  = (unsigned int*)(xw + (size_t)n * OUT_SIZE);
  float*        dinv = (float*)(deg + n);

  k_init_deg<<<(n + 255) / 256, 256, 0, stream>>>(deg, n);
  k_count_deg<<<(E + 255) / 256, 256, 0, stream>>>(row, col, deg, E);
  k_dinv<<<(n + 255) / 256, 256, 0, stream>>>(deg, dinv, n);

  const int tiles = (n + 15) / 16;
  k_gemm<<<(tiles + 7) / 8, 256, 0, stream>>>(x, W, xw, n);

  const size_t tot = (size_t)n * OUT_SIZE;
  k_self<<<(unsigned)((tot + 255) / 256), 256, 0, stream>>>(xw, dinv, out, n);
  k_scatter<<<(E + 3) / 4, 256, 0, stream>>>(row, col, dinv, xw, out, E);
}